// E2ESeqHead_76218489635043
// MI455X (gfx1250) — compile-verified
//
#include <hip/hip_runtime.h>
#include <hip/hip_bf16.h>
#include <stdint.h>

#define NB 16            // batch
#define NCLS 10
#define HW (512*512)     // 262144
#define EPB (NCLS*HW)    // 2,621,440 logits per batch
#define TOPK 500
#define NBINS 2048
#define CAP 8192
#define BLOCKS_PER_B 160
#define FLOATS_PER_BLOCK 16384   // 160 * 16384 = EPB
#define ITERS 16                 // 16 iters * 256 thr * 4 floats = 16384
#define NWAVES 8                 // 256 threads / wave32

typedef float v4f __attribute__((ext_vector_type(4)));

// Monotone order-preserving float -> u32 key (larger float => larger key).
__device__ __forceinline__ uint32_t order_key(float f) {
    uint32_t u = __float_as_uint(f);
    return u ^ ((uint32_t)((int32_t)u >> 31) | 0x80000000u);
}

// CDNA5 async global -> LDS copy (16B per lane), tracked by ASYNCcnt.
// th:TH_LOAD_HT keeps the logits resident in L2 (192MB) for the second pass.
__device__ __forceinline__ void async_copy16(uint32_t lds_byte, const float* gaddr) {
    asm volatile("global_load_async_to_lds_b128 %0, %1, off th:TH_LOAD_HT"
                 :: "v"(lds_byte), "v"((unsigned long long)(uintptr_t)gaddr)
                 : "memory");
}

// ---------------------------------------------------------------------------
// Pass A: streaming histogram of logit keys (top 11 bits). Data staged through
// LDS with double-buffered async copies; per-wave privatized histograms to
// kill inter-wave ds_add contention on hot bins.
// ---------------------------------------------------------------------------
__global__ void __launch_bounds__(256) k_hist(const float* __restrict__ logits,
                                              uint32_t* __restrict__ hist) {
    __shared__ uint32_t lh[NWAVES][NBINS];   // 64 KiB
    __shared__ float4 stage[2][256];         // 8 KiB
    const int t = threadIdx.x;
    const int w = t >> 5;                    // wave id (wave32)
    const int b = blockIdx.x / BLOCKS_PER_B;
    const int chunk = blockIdx.x % BLOCKS_PER_B;

    for (int i = t; i < NWAVES * NBINS; i += 256) lh[0][i] = 0u;
    __syncthreads();

    const float* base = logits + (size_t)b * EPB + (size_t)chunk * FLOATS_PER_BLOCK;
    const uint32_t lds0 = (uint32_t)(uintptr_t)&stage[0][t];
    const uint32_t lds1 = (uint32_t)(uintptr_t)&stage[1][t];

    async_copy16(lds0, base + t * 4);                       // prologue -> buf0
    for (int it = 0; it < ITERS; ++it) {
        if (it + 1 < ITERS) {
            async_copy16((it & 1) ? lds0 : lds1, base + (it + 1) * 1024 + t * 4);
            asm volatile("s_wait_asynccnt 1" ::: "memory"); // oldest copy done
        } else {
            asm volatile("s_wait_asynccnt 0" ::: "memory");
        }
        float4 v = stage[it & 1][t];
        atomicAdd(&lh[w][order_key(v.x) >> 21], 1u);
        atomicAdd(&lh[w][order_key(v.y) >> 21], 1u);
        atomicAdd(&lh[w][order_key(v.z) >> 21], 1u);
        atomicAdd(&lh[w][order_key(v.w) >> 21], 1u);
    }
    __syncthreads();
    for (int i = t; i < NBINS; i += 256) {
        uint32_t c = 0;
#pragma unroll
        for (int ww = 0; ww < NWAVES; ++ww) c += lh[ww][i];
        if (c) atomicAdd(&hist[b * NBINS + i], c);
    }
}

__global__ void __launch_bounds__(256) k_zero(uint32_t* hist, uint32_t* cnt) {
    int i = blockIdx.x * 256 + threadIdx.x;
    if (i < NB * NBINS) hist[i] = 0u;
    if (i < NB) cnt[i] = 0u;
}

// ---------------------------------------------------------------------------
// Threshold bin per batch = largest bin index whose suffix-sum >= TOPK.
// One block per batch: LDS suffix-scan (Hillis-Steele, 11 steps) + atomicMax.
// ---------------------------------------------------------------------------
__global__ void __launch_bounds__(256) k_thresh(const uint32_t* __restrict__ hist,
                                                uint32_t* __restrict__ thr) {
    __shared__ uint32_t s[NBINS];
    __shared__ int best;
    const int b = blockIdx.x;
    const int t = threadIdx.x;
    for (int i = t; i < NBINS; i += 256) s[i] = hist[b * NBINS + i];
    if (t == 0) best = 0;
    __syncthreads();

    for (int off = 1; off < NBINS; off <<= 1) {
        uint32_t tmp[NBINS / 256];
#pragma unroll
        for (int r = 0; r < NBINS / 256; ++r) {
            int i = t + r * 256;
            tmp[r] = (i + off < NBINS) ? s[i + off] : 0u;
        }
        __syncthreads();
#pragma unroll
        for (int r = 0; r < NBINS / 256; ++r) s[t + r * 256] += tmp[r];
        __syncthreads();
    }

    int local = -1;
#pragma unroll
    for (int r = 0; r < NBINS / 256; ++r) {
        int i = t + r * 256;
        if (s[i] >= TOPK && i > local) local = i;
    }
    if (local >= 0) atomicMax(&best, local);
    __syncthreads();
    if (t == 0) thr[b] = (uint32_t)best;
}

// ---------------------------------------------------------------------------
// Pass B: append every element with bin >= threshold bin to candidate buffer.
// Non-temporal reads: this is the last streaming pass over the logits.
// ---------------------------------------------------------------------------
__global__ void __launch_bounds__(256) k_scatter(const float* __restrict__ logits,
                                                 const uint32_t* __restrict__ thr,
                                                 uint32_t* __restrict__ cnt,
                                                 uint2* __restrict__ cand) {
    const int t = threadIdx.x;
    const int b = blockIdx.x / BLOCKS_PER_B;
    const int chunk = blockIdx.x % BLOCKS_PER_B;
    const uint32_t tb = thr[b];
    const float* base = logits + (size_t)b * EPB + (size_t)chunk * FLOATS_PER_BLOCK;
    const uint32_t e0 = (uint32_t)chunk * FLOATS_PER_BLOCK;

    for (int it = 0; it < ITERS; ++it) {
        const int off = it * 1024 + t * 4;
        v4f v = __builtin_nontemporal_load((const v4f*)(base + off));
        uint32_t key[4] = { order_key(v.x), order_key(v.y), order_key(v.z), order_key(v.w) };
#pragma unroll
        for (int j = 0; j < 4; ++j) {
            if ((key[j] >> 21) >= tb) {
                uint32_t pos = atomicAdd(&cnt[b], 1u);
                if (pos < CAP) cand[(size_t)b * CAP + pos] = make_uint2(key[j], e0 + off + j);
            }
        }
    }
}

// ---------------------------------------------------------------------------
// Final: per-batch bitonic sort (key desc, idx asc -> deterministic) of
// candidates in LDS, then emit top-500: decoded box (7), full 10-class
// sigmoid score vector, label = class+1.
// ---------------------------------------------------------------------------
__global__ void __launch_bounds__(512) k_final(const float* __restrict__ logits,
                                               const float* __restrict__ pboxes,
                                               const uint32_t* __restrict__ cnt,
                                               const uint2* __restrict__ cand,
                                               float* __restrict__ out) {
    __shared__ uint32_t skey[CAP];
    __shared__ uint32_t sidx[CAP];
    const int b = blockIdx.x;
    const int t = threadIdx.x;
    uint32_t n = cnt[b];
    if (n > CAP) n = CAP;

    for (int i = t; i < CAP; i += 512) {
        if (i < (int)n) {
            uint2 e = cand[(size_t)b * CAP + i];
            skey[i] = e.x; sidx[i] = e.y;
        } else {
            skey[i] = 0u; sidx[i] = 0xFFFFFFFFu;
        }
    }
    __syncthreads();

    for (int kk = 2; kk <= CAP; kk <<= 1) {
        for (int jj = kk >> 1; jj > 0; jj >>= 1) {
            for (int i = t; i < CAP; i += 512) {
                int ixj = i ^ jj;
                if (ixj > i) {
                    uint32_t a = skey[i], c = skey[ixj];
                    uint32_t ia = sidx[i], ic = sidx[ixj];
                    bool before = (a > c) || (a == c && ia < ic); // (key desc, idx asc)
                    bool dir = ((i & kk) == 0);                   // descending region
                    if (before != dir) {
                        skey[i] = c; skey[ixj] = a;
                        sidx[i] = ic; sidx[ixj] = ia;
                    }
                }
            }
            __syncthreads();
        }
    }

    const float* lg = logits + (size_t)b * EPB;
    const float* pb = pboxes + (size_t)b * 8 * HW;
    float* boxes_out  = out;
    float* scores_out = out + (size_t)NB * TOPK * 7;
    float* labels_out = out + (size_t)NB * TOPK * 7 + (size_t)NB * TOPK * NCLS;

    if (t < TOPK) {
        uint32_t idx = (t < (int)n) ? sidx[t] : 0u;
        uint32_t c   = idx / HW;
        uint32_t loc = idx % HW;
        uint32_t col = loc & 511u;
        uint32_t row = loc >> 9;

        float p0 = pb[0 * HW + loc], p1 = pb[1 * HW + loc], p2 = pb[2 * HW + loc];
        float p3 = pb[3 * HW + loc], p4 = pb[4 * HW + loc], p5 = pb[5 * HW + loc];
        float p6 = pb[6 * HW + loc], p7 = pb[7 * HW + loc];

        float x = fminf(fmaxf(p0, -0.5f), 0.5f) * 0.2f + ((col + 0.5f) * 0.2f - 51.2f);
        float y = fminf(fmaxf(p1, -0.5f), 0.5f) * 0.2f + ((row + 0.5f) * 0.2f - 51.2f);

        float* bo = boxes_out + ((size_t)b * TOPK + t) * 7;
        bo[0] = x; bo[1] = y; bo[2] = p2;
        bo[3] = expf(p3); bo[4] = expf(p4); bo[5] = expf(p5);
        bo[6] = atan2f(p6, p7);

        float* so = scores_out + ((size_t)b * TOPK + t) * NCLS;
#pragma unroll
        for (int cc = 0; cc < NCLS; ++cc) {
            float z = lg[(size_t)cc * HW + loc];
            so[cc] = 1.0f / (1.0f + expf(-z));
        }
        labels_out[(size_t)b * TOPK + t] = (float)(c + 1u);
    }
}

extern "C" void kernel_launch(void* const* d_in, const int* in_sizes, int n_in,
                              void* d_out, int out_size, void* d_ws, size_t ws_size,
                              hipStream_t stream) {
    const float* pboxes = (const float*)d_in[0];   // (16, 8, 512, 512)
    const float* logits = (const float*)d_in[1];   // (16, 10, 512, 512)
    float* out = (float*)d_out;                    // boxes(56000) | scores(80000) | labels(8000)

    uint8_t* ws = (uint8_t*)d_ws;
    uint2*    cand = (uint2*)ws;                                   // 16*8192*8 = 1 MiB
    uint32_t* hist = (uint32_t*)(ws + (size_t)NB * CAP * sizeof(uint2));
    uint32_t* cnt  = hist + NB * NBINS;
    uint32_t* thr  = cnt + NB;

    k_zero   <<<(NB * NBINS + 255) / 256, 256, 0, stream>>>(hist, cnt);
    k_hist   <<<NB * BLOCKS_PER_B, 256, 0, stream>>>(logits, hist);
    k_thresh <<<NB, 256, 0, stream>>>(hist, thr);
    k_scatter<<<NB * BLOCKS_PER_B, 256, 0, stream>>>(logits, thr, cnt, cand);
    k_final  <<<NB, 512, 0, stream>>>(logits, pboxes, cnt, cand, out);
}